// CBFLayer_16432544874701
// MI455X (gfx1250) — compile-verified
//
#include <hip/hip_runtime.h>

// CBF-QP batch solver, B = 4,194,304 rows of 2-D controls.
// Pass 1: analytic cases (clip / halfplane projection); hard rows -> worklist
//         (block-aggregated atomics: LDS rank + one global atomicAdd/block).
// Pass 2: dense root-find (doubling + bisection + Newton) on worklist only,
//         with wave-vote early exit at per-lane fixed points (bit-exact vs.
//         the reference's fixed 40/60 iteration counts).

#define EPSV     1e-12f
#define TOLV     1e-6f
#define LAM_F    10000.0f
#define LAM_INV  (1.0f / 10000.0f)   // fl(1e-4), <=1ulp vs true divide

#if defined(__has_builtin)
#if __has_builtin(__builtin_amdgcn_global_load_async_to_lds_b128) && \
    __has_builtin(__builtin_amdgcn_s_wait_asynccnt)
#define USE_ASYNC_LDS 1
#endif
#endif

#ifdef USE_ASYNC_LDS
typedef __attribute__((__vector_size__(4 * sizeof(int)))) int v4i_t;
typedef __attribute__((address_space(1))) v4i_t* g_v4i_p;   // global
typedef __attribute__((address_space(3))) v4i_t* l_v4i_p;   // LDS
#endif

// phi(t) = (p - t*A) - (b + t/LAM) * sqrt(max(N - 2 t p + t^2 A, eps))  (U_MAX=1)
__device__ __forceinline__ float phi_eval(float t, float A, float m2p, float p,
                                          float N, float b) {
    float q   = fmaf(t * t, A, fmaf(t, m2p, N));
    float nrm = __builtin_sqrtf(fmaxf(q, EPSV));
    return fmaf(-(b + t * LAM_INV), nrm, fmaf(-t, A, p));
}

__device__ __forceinline__ void solve_u3(float ux, float uy, float ax, float ay,
                                         float b, float A, float p, float N,
                                         float &ox, float &oy) {
    const float m2p = -2.0f * p;

    // Doubling: once phi(t_hi) <= 0 the lane's t_hi is a fixed point forever.
    float t_hi = 1.0f;
    for (int i = 0; i < 40; ++i) {
        bool pos = phi_eval(t_hi, A, m2p, p, N, b) > 0.0f;
        if (!__any((int)pos)) break;
        t_hi = pos ? (t_hi * 2.0f) : t_hi;
    }

    // Bisection: an iteration changing neither lo nor hi can never change
    // anything again (deterministic map) -> break at the wave fixed point.
    float lo = 0.0f, hi = t_hi;
    for (int i = 0; i < 60; ++i) {
        float mid = 0.5f * (lo + hi);
        bool  pos = phi_eval(mid, A, m2p, p, N, b) > 0.0f;
        float nlo = pos ? mid : lo;
        float nhi = pos ? hi  : mid;
        bool  chg = (nlo != lo) | (nhi != hi);
        lo = nlo;
        hi = nhi;
        if (!__any((int)chg)) break;
    }
    float t3 = 0.5f * (lo + hi);

    // 3 Newton polish steps (as reference)
#pragma unroll
    for (int i = 0; i < 3; ++i) {
        float q   = fmaf(t3 * t3, A, fmaf(t3, m2p, N));
        float nrm = __builtin_sqrtf(fmaxf(q, EPSV));
        float bt  = b + t3 * LAM_INV;
        float pm  = fmaf(-t3, A, p);                       // p - t3*A
        float f   = fmaf(-bt, nrm, pm);
        float df  = -A - nrm * LAM_INV + bt * pm / nrm;    // -(t3*A - p) = pm
        df = (fabsf(df) > 1e-8f) ? df : -1e-8f;
        t3 = t3 - f / df;
    }
    float q    = fmaf(t3 * t3, A, fmaf(t3, m2p, N));
    float nrm3 = __builtin_sqrtf(fmaxf(q, EPSV));
    float k3   = fmaxf(nrm3, 1.0f);                        // U_MAX = 1
    float ik3  = 1.0f / k3;
    ox = fmaf(-t3, ax, ux) * ik3;
    oy = fmaf(-t3, ay, uy) * ik3;
}

template <bool USE_WL>
__global__ __launch_bounds__(256)
void cbf_pass1(const float* __restrict__ u_nom, const float* __restrict__ obs,
               float* __restrict__ out, unsigned int* __restrict__ wl_count,
               unsigned int* __restrict__ wl, int n) {
    __shared__ __align__(16) float s_obs[256 * 6];
    __shared__ unsigned int s_cnt;
    __shared__ unsigned int s_base;

    const int block0 = blockIdx.x * 256;
    if (USE_WL && threadIdx.x == 0) s_cnt = 0u;

    // Stage this block's obs tile into LDS with coalesced 16B transfers.
    {
        const float4* g4     = reinterpret_cast<const float4*>(obs);
        float4*       s4     = reinterpret_cast<float4*>(s_obs);
        const size_t  base4  = (size_t)block0 * 6u / 4u;   // block0 multiple of 256
        const size_t  total4 = (size_t)n * 6u / 4u;
#ifdef USE_ASYNC_LDS
        // gfx1250 direct global->LDS async copy (ASYNCcnt-tracked), no VGPR round trip.
        for (int i = threadIdx.x; i < 384; i += 256) {
            size_t gi = base4 + (size_t)i;
            if (gi < total4) {
                __builtin_amdgcn_global_load_async_to_lds_b128(
                    (g_v4i_p)(g4 + gi), (l_v4i_p)(s4 + i), 0, 0);
            }
        }
        __builtin_amdgcn_s_wait_asynccnt(0);
#else
        for (int i = threadIdx.x; i < 384; i += 256) {
            size_t gi = base4 + (size_t)i;
            if (gi < total4) s4[i] = g4[gi];
        }
#endif
    }
    // Prefetch next tile while we compute (gfx1250 global_prefetch_b8).
    if (block0 + 256 < n) {
        __builtin_prefetch(obs + (size_t)(block0 + 256) * 6u, 0, 3);
        __builtin_prefetch(u_nom + (size_t)(block0 + 256) * 2u, 0, 3);
    }
    __syncthreads();   // covers tile staging and s_cnt init

    const int  row   = block0 + (int)threadIdx.x;
    const bool valid = (row < n);
    bool hard = false;

    if (valid) {
        const float2 u  = reinterpret_cast<const float2*>(u_nom)[row];
        const float2 pr = *reinterpret_cast<const float2*>(&s_obs[threadIdx.x * 6 + 2]);
        const float2 vh = *reinterpret_cast<const float2*>(&s_obs[threadIdx.x * 6 + 4]);

        const float px = pr.x, py = pr.y, vx = vh.x, vy = vh.y;
        const float h  = fmaf(px, px, py * py) - 1.0f;               // SAFE_DIST^2 = 1
        const float ax = -2.0f * px, ay = -2.0f * py;
        const float b  = 2.0f * h - 2.0f * fmaf(px, vx, py * vy);    // ALPHA = 2
        const float A  = fmaf(ax, ax, ay * ay);
        const float p  = fmaf(ax, u.x, ay * u.y);
        const float N  = fmaf(u.x, u.x, u.y * u.y);

        // Case 1: norm-clip feasibility
        const float nn  = __builtin_sqrtf(fmaxf(N, EPSV));
        const float s1  = fminf(1.0f, 1.0f / nn);                    // U_MAX = 1
        const float u1x = u.x * s1, u1y = u.y * s1;
        const bool  feas1 = (fmaf(ax, u1x, ay * u1y) <= b + TOLV);

        // Case 2: halfplane projection
        const float t2  = LAM_F * (p - b) / fmaf(LAM_F, A, 1.0f);
        const float u2x = fmaf(-t2, ax, u.x), u2y = fmaf(-t2, ay, u.y);
        const bool  ok2 = (t2 >= -TOLV) && (fmaf(u2x, u2x, u2y * u2y) <= 1.0f + TOLV);

        float ox, oy;
        bool  have = true;
        if (feas1)    { ox = u1x; oy = u1y; }
        else if (ok2) { ox = u2x; oy = u2y; }
        else if (USE_WL) { hard = true; have = false; ox = 0.0f; oy = 0.0f; }
        else          { solve_u3(u.x, u.y, ax, ay, b, A, p, N, ox, oy); }

        if (have) reinterpret_cast<float2*>(out)[row] = make_float2(ox, oy);
    }

    if (USE_WL) {
        // Block-aggregated worklist append: LDS rank + 1 global atomic per block.
        unsigned int lidx = 0u;
        if (hard) lidx = atomicAdd(&s_cnt, 1u);
        __syncthreads();
        if (threadIdx.x == 0 && s_cnt != 0u) s_base = atomicAdd(wl_count, s_cnt);
        __syncthreads();
        if (hard) wl[s_base + lidx] = (unsigned int)row;
    }
}

__global__ __launch_bounds__(256)
void cbf_pass2(const float* __restrict__ u_nom, const float* __restrict__ obs,
               float* __restrict__ out, const unsigned int* __restrict__ wl_count,
               const unsigned int* __restrict__ wl) {
    const unsigned int cnt = *wl_count;
    const unsigned int i   = blockIdx.x * 256u + threadIdx.x;
    if (i >= cnt) return;

    const int row = (int)wl[i];
    const float2 u  = reinterpret_cast<const float2*>(u_nom)[row];
    const float2 pr = *reinterpret_cast<const float2*>(obs + (size_t)row * 6u + 2u);
    const float2 vh = *reinterpret_cast<const float2*>(obs + (size_t)row * 6u + 4u);

    const float px = pr.x, py = pr.y, vx = vh.x, vy = vh.y;
    const float h  = fmaf(px, px, py * py) - 1.0f;
    const float ax = -2.0f * px, ay = -2.0f * py;
    const float b  = 2.0f * h - 2.0f * fmaf(px, vx, py * vy);
    const float A  = fmaf(ax, ax, ay * ay);
    const float p  = fmaf(ax, u.x, ay * u.y);
    const float N  = fmaf(u.x, u.x, u.y * u.y);

    float ox, oy;
    solve_u3(u.x, u.y, ax, ay, b, A, p, N, ox, oy);
    reinterpret_cast<float2*>(out)[row] = make_float2(ox, oy);
}

extern "C" void kernel_launch(void* const* d_in, const int* in_sizes, int n_in,
                              void* d_out, int out_size, void* d_ws, size_t ws_size,
                              hipStream_t stream) {
    const float* u_nom = (const float*)d_in[0];
    const float* obs   = (const float*)d_in[1];
    float*       out   = (float*)d_out;

    const int n = in_sizes[0] / 2;                 // rows
    if (n <= 0) return;
    const int blocks = (n + 255) / 256;

    unsigned int* wl_count = (unsigned int*)d_ws;
    unsigned int* wl       = wl_count + 4;         // worklist starts at +16 B
    const bool use_wl = (ws_size >= 16u + (size_t)n * sizeof(unsigned int));

    if (use_wl) {
        (void)hipMemsetAsync(d_ws, 0, 16, stream); // zero counter (graph-capturable)
        cbf_pass1<true><<<blocks, 256, 0, stream>>>(u_nom, obs, out, wl_count, wl, n);
        cbf_pass2<<<blocks, 256, 0, stream>>>(u_nom, obs, out, wl_count, wl);
    } else {
        cbf_pass1<false><<<blocks, 256, 0, stream>>>(u_nom, obs, out, wl_count, wl, n);
    }
}